// MultiHeadAttention_48919677502060
// MI455X (gfx1250) — compile-verified
//
#include <hip/hip_runtime.h>
#include <cstdint>
#include <cstddef>

// ---- problem constants (from reference) ----
#define B_ 2
#define L_ 2048
#define C_ 1024
#define H_ 16
#define D_ 64
#define N_ (B_ * L_)          // 4096 total rows

typedef __attribute__((ext_vector_type(2))) float v2f;
typedef __attribute__((ext_vector_type(8))) float v8f;
typedef __attribute__((ext_vector_type(4))) unsigned int v4u;
typedef __attribute__((ext_vector_type(4))) int v4i;
typedef __attribute__((ext_vector_type(8))) int v8i;

#if defined(__has_builtin)
#if __has_builtin(__builtin_amdgcn_tensor_load_to_lds) && \
    __has_builtin(__builtin_amdgcn_s_wait_tensorcnt)
#define USE_TDM 1
#endif
#endif

// D = A(16x4) * B(4x16) + C(16x16), f32, wave32.
// A frag: lane holds A[lane&15][(lane>>4)*2 + {0,1}]   (ISA 32-bit A layout)
// B frag: lane holds B[(lane>>4)*2 + {0,1}][lane&15]
// C/D  : vgpr r, lanes 0-15: M=r; lanes 16-31: M=r+8; N = lane&15
static __device__ __forceinline__ v8f wmma_f32(v2f a, v2f b, v8f c) {
    return __builtin_amdgcn_wmma_f32_16x16x4_f32(false, a, false, b, (short)0, c,
                                                 false, false);
}

#ifdef USE_TDM
// TDM: contiguous 4KB load (16 rows x 64 f32) into LDS with padding of
// 4 DWORDs after every 64 DWORDs -> LDS row stride 68 floats (bank-conflict
// free fragment reads). D# per CDNA5 ISA.
static __device__ __forceinline__ void tdm_load_tile(const float* gptr,
                                                     unsigned lds_off) {
    const unsigned long long ga = (unsigned long long)(uintptr_t)gptr;
    v4u g0;
    g0.x = 1u;                                     // count=1, user mode
    g0.y = lds_off;                                // lds_addr (bytes)
    g0.z = (unsigned)(ga & 0xFFFFFFFFu);           // global_addr[31:0]
    g0.w = (unsigned)((ga >> 32) & 0x01FFFFFFu)    // global_addr[56:32]
           | (2u << 30);                           // type = 2 ("image")
    v8i g1;
    // data_size=2 (4B) | pad_enable | pad_interval=5 (64 dw) | pad_amount=3 (4 dw)
    g1[0] = 0x07520000;
    g1[1] = (int)(1024u << 16);   // tensor_dim0 = 1024 dwords (bits 63:48)
    g1[2] = (int)(1u << 16);      // tensor_dim1 = 1          (bits 95:80)
    g1[3] = (int)(1024u << 16);   // tile_dim0   = 1024       (bits 127:112)
    g1[4] = 0;                    // tile_dim1/2 unused
    g1[5] = 1024;                 // tensor_dim0_stride[31:0]
    g1[6] = 0;
    g1[7] = 0;
    v4i z4 = {0, 0, 0, 0};
#if defined(__clang_major__) && __clang_major__ >= 23
    v8i z8 = {0, 0, 0, 0, 0, 0, 0, 0};
    __builtin_amdgcn_tensor_load_to_lds(g0, g1, z4, z4, z8, 0);
#else
    __builtin_amdgcn_tensor_load_to_lds(g0, g1, z4, z4, 0);
#endif
}
#endif  // USE_TDM

// ============================================================================
// Kernel 1: Q/K/V projection (y = x @ W^T) + RoPE, output layout [B,H,L,D].
// grid = (N/16, 32, 3): x = row tile, y = (head, pair) with pair columns
// o0 = h*64 + p*16 (tileA) and o0+32 (tileB) so rotate-half partners are
// in-wave; z selects Wq/Wk/Wv. One wave (32 threads) per block.
// ============================================================================
__global__ __launch_bounds__(32) void qkv_rope_kernel(
    const float* __restrict__ x,
    const float* __restrict__ Wq, const float* __restrict__ Wk,
    const float* __restrict__ Wv,
    float* __restrict__ outq, float* __restrict__ outk, float* __restrict__ outv)
{
    const int lane = threadIdx.x;
    const int lm = lane & 15;            // tile row (A) / tile col (B,C)
    const int kg = (lane >> 4) * 2;      // K sub-offset: 0 or 2
    const int hi = (lane >> 4) * 8;      // C-layout row offset: 0 or 8

    const int n0 = blockIdx.x * 16;
    const int h  = blockIdx.y >> 1;
    const int p  = blockIdx.y & 1;
    const int o0 = h * 64 + p * 16;      // tileA cols; tileB cols = o0+32
    const int z  = blockIdx.z;           // 0=Q 1=K 2=V

    const float* W   = (z == 0) ? Wq : ((z == 1) ? Wk : Wv);
    float*       out = (z == 0) ? outq : ((z == 1) ? outk : outv);

    const float* xrow  = x + (size_t)(n0 + lm) * C_;
    const float* wArow = W + (size_t)(o0 + lm) * C_;
    const float* wBrow = W + (size_t)(o0 + 32 + lm) * C_;

    v8f accA = {};
    v8f accB = {};
    for (int kb = 0; kb < C_; kb += 64) {
        if (kb + 64 < C_) {
            __builtin_prefetch(xrow  + kb + 64, 0, 1);   // global_prefetch_b8
            __builtin_prefetch(wArow + kb + 64, 0, 1);
            __builtin_prefetch(wBrow + kb + 64, 0, 1);
        }
#pragma unroll
        for (int k0 = kb; k0 < kb + 64; k0 += 4) {
            v2f a  = *(const v2f*)(xrow  + k0 + kg);
            v2f bA = *(const v2f*)(wArow + k0 + kg);
            v2f bB = *(const v2f*)(wBrow + k0 + kg);
            accA = wmma_f32(a, bA, accA);
            accB = wmma_f32(a, bB, accB);
        }
    }

    // RoPE: tileA column d = p*16+lm in [0,32); partner column d+32 in tileB.
    // inv_freq[d] = theta^(-d/32); both partners share the same angle.
    const int dA = p * 16 + lm;                            // 0..31
    const float inv_f = __expf(-0.33811807f * (float)dA);  // ln(50000)/32

#pragma unroll
    for (int r = 0; r < 8; ++r) {
        const int n = n0 + r + hi;                    // global row
        const int b = n >> 11;                        // n / 2048
        const int l = n & (L_ - 1);
        float va = accA[r];
        float vb = accB[r];
        float ra = va, rb = vb;
        if (z < 2) {                                  // uniform per block
            const float fr = (float)l * inv_f;
            float s, c;
            sincosf(fr, &s, &c);                      // libm: full range reduction
            ra = va * c - vb * s;                     // q*cos + rot_half(q)*sin
            rb = vb * c + va * s;
        }
        float* dst = out + (((size_t)(b * H_ + h) * L_ + l) * D_);
        dst[dA]      = ra;
        dst[dA + 32] = rb;
    }
}

// ============================================================================
// Kernel 2: flash-style masked attention, one 16-row query tile per wave.
// grid = (B*H, L/16/4), block = 128 (4 waves, all on the SAME (b,h)).
// K and V tiles are shared block-wide: wave 0 streams both via the Tensor
// Data Mover (double-buffered, s_wait_tensorcnt(2) covers the current pair,
// in-order per wave), barriers publish the buffers. TDM LDS padding gives a
// 68-float row stride -> conflict-free ds reads for both K B-frags (row=s,
// scan k) and V B-frags (row=k, lanes scan consecutive d).
// Online softmax via 16-lane shuffle reductions; P staged through padded LDS
// to convert C-layout -> A-layout for the P@V WMMAs.
// ============================================================================
__global__ __launch_bounds__(128) void attn_kernel(
    const float* __restrict__ Q, const float* __restrict__ K,
    const float* __restrict__ V, const unsigned char* __restrict__ mask,
    float* __restrict__ O)
{
    __shared__ float plds[4][16 * 20];   // P stage: stride 20, conflict-free
#ifdef USE_TDM
    __shared__ float kbuf[2][1088];      // block-shared double-buffered K tile
    __shared__ float vbuf[2][1088];      // block-shared double-buffered V tile
#endif

    const int lane = threadIdx.x & 31;
    const int wave = threadIdx.x >> 5;
    const int lm = lane & 15;
    const int kg = (lane >> 4) * 2;
    const int hi = (lane >> 4) * 8;

    const int bh = blockIdx.x;                 // 0..B*H-1
    const int b = bh >> 4, h = bh & 15;
    const int l0 = (blockIdx.y * 4 + wave) * 16;

    const size_t base = (size_t)(b * H_ + h) * L_ * D_;
    const float* Qbh = Q + base;
    const float* Kbh = K + base;
    const float* Vbh = V + base;
    const unsigned char* mrow = mask + (size_t)b * L_;
    float* pl = plds[wave];

    // Preload Q A-fragments for all 16 K-steps (D = 64 = 16 * 4).
    v2f qa[16];
    const float* qrow = Qbh + (size_t)(l0 + lm) * D_;
#pragma unroll
    for (int ks = 0; ks < 16; ++ks)
        qa[ks] = *(const v2f*)(qrow + ks * 4 + kg);

    v8f oa0 = {}, oa1 = {}, oa2 = {}, oa3 = {};
    float rmax[8], rsum[8];
#pragma unroll
    for (int r = 0; r < 8; ++r) { rmax[r] = -1e30f; rsum[r] = 0.0f; }
    const float scale = 0.125f;   // 1/sqrt(64)
    const int NT = L_ / 16;

#ifdef USE_TDM
    const unsigned klo[2] = { (unsigned)(uintptr_t)&kbuf[0][0],
                              (unsigned)(uintptr_t)&kbuf[1][0] };
    const unsigned vlo[2] = { (unsigned)(uintptr_t)&vbuf[0][0],
                              (unsigned)(uintptr_t)&vbuf[1][0] };
    if (wave == 0) {                       // prefetch tile 0 (K then V)
        tdm_load_tile(Kbh, klo[0]);
        tdm_load_tile(Vbh, vlo[0]);
    }
#endif

    for (int st = 0; st < NT; ++st) {
        const int s0 = st * 16;

#ifdef USE_TDM
        if (wave == 0) {
            if (st + 1 < NT) {             // issue next pair, cover current
                const float* nx = Kbh + (size_t)(s0 + 16) * D_;
                tdm_load_tile(nx, klo[(st + 1) & 1]);
                tdm_load_tile(Vbh + (size_t)(s0 + 16) * D_, vlo[(st + 1) & 1]);
                __builtin_amdgcn_s_wait_tensorcnt(2);
            } else {
                __builtin_amdgcn_s_wait_tensorcnt(0);
            }
        }
        __syncthreads();                   // current K/V tiles valid block-wide

        // ---- scores: S = Q_tile @ K_tile^T  (B frag: B[k][n] = K[s0+n][k])
        v8f sc = {};
        const float* kb = &kbuf[st & 1][0];        // padded rows: stride 68
#pragma unroll
        for (int ks = 0; ks < 16; ++ks) {
            v2f bf = *(const v2f*)(kb + lm * 68 + ks * 4 + kg);
            sc = wmma_f32(qa[ks], bf, sc);
        }
#else
        v8f sc = {};
        const float* krow = Kbh + (size_t)(s0 + lm) * D_;
#pragma unroll
        for (int ks = 0; ks < 16; ++ks) {
            v2f bf = *(const v2f*)(krow + ks * 4 + kg);
            sc = wmma_f32(qa[ks], bf, sc);
        }
#endif

        const float bias = mrow[s0 + lm] ? 0.0f : -1e30f;  // column = key pos

        // ---- online softmax (rows live across a 16-lane group)
        float corr[8];
#pragma unroll
        for (int r = 0; r < 8; ++r) {
            float v = sc[r] * scale + bias;
            float mx = v;
            mx = fmaxf(mx, __shfl_xor(mx, 1, 32));
            mx = fmaxf(mx, __shfl_xor(mx, 2, 32));
            mx = fmaxf(mx, __shfl_xor(mx, 4, 32));
            mx = fmaxf(mx, __shfl_xor(mx, 8, 32));
            const float nm = fmaxf(rmax[r], mx);
            const float pv = __expf(v - nm);
            float su = pv;
            su += __shfl_xor(su, 1, 32);
            su += __shfl_xor(su, 2, 32);
            su += __shfl_xor(su, 4, 32);
            su += __shfl_xor(su, 8, 32);
            corr[r] = __expf(rmax[r] - nm);
            rmax[r] = nm;
            rsum[r] = rsum[r] * corr[r] + su;
            pl[(r + hi) * 20 + lm] = pv;       // stage P[m][n] in LDS
        }
#pragma unroll
        for (int r = 0; r < 8; ++r) {
            oa0[r] *= corr[r]; oa1[r] *= corr[r];
            oa2[r] *= corr[r]; oa3[r] *= corr[r];
        }

        // ---- O += P @ V_tile  (A from LDS in A-layout, B cols = d chunk)
#pragma unroll
        for (int ks = 0; ks < 4; ++ks) {
            v2f af;
            af.x = pl[lm * 20 + ks * 4 + kg];
            af.y = pl[lm * 20 + ks * 4 + kg + 1];
#ifdef USE_TDM
            const float* vb = &vbuf[st & 1][(ks * 4 + kg) * 68 + lm];
            v2f b0; b0.x = vb[0];  b0.y = vb[68 + 0];
            v2f b1; b1.x = vb[16]; b1.y = vb[68 + 16];
            v2f b2; b2.x = vb[32]; b2.y = vb[68 + 32];
            v2f b3; b3.x = vb[48]; b3.y = vb[68 + 48];
#else
            const float* vb = Vbh + (size_t)(s0 + ks * 4 + kg) * D_ + lm;
            v2f b0; b0.x = vb[0];  b0.y = vb[D_ + 0];
            v2f b1; b1.x = vb[16]; b1.y = vb[D_ + 16];
            v2f b2; b2.x = vb[32]; b2.y = vb[D_ + 32];
            v2f b3; b3.x = vb[48]; b3.y = vb[D_ + 48];
#endif
            oa0 = wmma_f32(af, b0, oa0);
            oa1 = wmma_f32(af, b1, oa1);
            oa2 = wmma_f32(af, b2, oa2);
            oa3 = wmma_f32(af, b3, oa3);
        }

#ifdef USE_TDM
        __syncthreads();   // all waves done reading before next TDM overwrite
#endif
    }

    // ---- normalize and store as [B, L, C] (row n, col h*64 + d)
    float* obase = O + ((size_t)(b * L_) + l0) * C_ + h * D_;
#pragma unroll
    for (int r = 0; r < 8; ++r) {
        const float inv = 1.0f / rsum[r];
        float* orow = obase + (size_t)(r + hi) * C_;
        orow[lm]      = oa0[r] * inv;
        orow[lm + 16] = oa1[r] * inv;
        orow[lm + 32] = oa2[r] * inv;
        orow[lm + 48] = oa3[r] * inv;
    }
}

// ============================================================================
// Kernel 3: output projection Y = A @ Wo^T + bo.  grid = (N/16, C/32): each
// wave does two 16-col tiles sharing the A fragment (2 WMMA per A load).
// ============================================================================
__global__ __launch_bounds__(32) void out_proj_kernel(
    const float* __restrict__ A, const float* __restrict__ Wo,
    const float* __restrict__ bo, float* __restrict__ Y)
{
    const int lane = threadIdx.x;
    const int lm = lane & 15;
    const int kg = (lane >> 4) * 2;
    const int hi = (lane >> 4) * 8;
    const int n0 = blockIdx.x * 16;
    const int o0 = blockIdx.y * 32;

    const float* arow  = A  + (size_t)(n0 + lm) * C_;
    const float* wArow = Wo + (size_t)(o0 + lm) * C_;
    const float* wBrow = Wo + (size_t)(o0 + 16 + lm) * C_;

    v8f accA = {};
    v8f accB = {};
    for (int kb = 0; kb < C_; kb += 64) {
        if (kb + 64 < C_) {
            __builtin_prefetch(arow  + kb + 64, 0, 1);
            __builtin_prefetch(wArow + kb + 64, 0, 1);
            __builtin_prefetch(wBrow + kb + 64, 0, 1);
        }
#pragma unroll
        for (int k0 = kb; k0 < kb + 64; k0 += 4) {
            v2f a  = *(const v2f*)(arow  + k0 + kg);
            v2f bA = *(const v2f*)(wArow + k0 + kg);
            v2f bB = *(const v2f*)(wBrow + k0 + kg);
            accA = wmma_f32(a, bA, accA);
            accB = wmma_f32(a, bB, accB);
        }
    }

    const float biasA = bo[o0 + lm];
    const float biasB = bo[o0 + 16 + lm];
#pragma unroll
    for (int r = 0; r < 8; ++r) {
        float* yrow = Y + (size_t)(n0 + r + hi) * C_ + o0;
        yrow[lm]      = accA[r] + biasA;
        yrow[lm + 16] = accB[r] + biasB;
    }
}

// ============================================================================
// Host launcher
// ============================================================================
extern "C" void kernel_launch(void* const* d_in, const int* in_sizes, int n_in,
                              void* d_out, int out_size, void* d_ws, size_t ws_size,
                              hipStream_t stream)
{
    (void)in_sizes; (void)n_in; (void)out_size; (void)ws_size;

    const float* x  = (const float*)d_in[0];
    // jax bool array -> 1 byte per element
    const unsigned char* mask = (const unsigned char*)d_in[1];
    const float* Wq = (const float*)d_in[2];
    const float* Wk = (const float*)d_in[3];
    const float* Wv = (const float*)d_in[4];
    const float* Wo = (const float*)d_in[5];
    const float* bo = (const float*)d_in[6];
    float* Y = (float*)d_out;

    // workspace: Q,K,V in [B,H,L,D] + attention output in [B,L,C] (4x16MB)
    const size_t BHLD = (size_t)B_ * H_ * L_ * D_;
    float* qh = (float*)d_ws;
    float* kh = qh + BHLD;
    float* vh = kh + BHLD;
    float* ao = vh + BHLD;

    dim3 g1(N_ / 16, 32, 3);
    qkv_rope_kernel<<<g1, 32, 0, stream>>>(x, Wq, Wk, Wv, qh, kh, vh);

    dim3 g2(B_ * H_, L_ / 16 / 4);
    attn_kernel<<<g2, 128, 0, stream>>>(qh, kh, vh, mask, ao);

    dim3 g3(N_ / 16, C_ / 32);
    out_proj_kernel<<<g3, 32, 0, stream>>>(ao, Wo, bo, Y);
}